// AttnDecoderRNN_70866960384401
// MI455X (gfx1250) — compile-verified
//
#include <hip/hip_runtime.h>
#include <math.h>

// Problem dims (fixed by the reference)
#define Bb 64
#define Ss 128
#define Hh 1024
#define Vv 32000
#define Tt 10
#define SOS 0

typedef float v2f __attribute__((ext_vector_type(2)));
typedef float v8f __attribute__((ext_vector_type(8)));

#define KT  32      // K tile
#define LDP 34      // padded LDS row stride (dwords) to break bank conflicts
#define ASZ (64 * LDP)
#define WSZ (128 * LDP)

// ---------------------------------------------------------------------------
// CDNA5 async global->LDS copy (ASYNCcnt-tracked, bypasses VGPRs).
// Each active lane copies 16 bytes from its global address to its LDS address.
// ---------------------------------------------------------------------------
__device__ __forceinline__ void async_copy_b128(float* lds_dst, const float* gsrc)
{
    asm volatile("global_load_async_to_lds_b128 %0, %1, off"
                 :
                 : "v"((__attribute__((address_space(3))) void*)lds_dst),
                   "v"(gsrc)
                 : "memory");
}

__device__ __forceinline__ void wait_async0()
{
    asm volatile("s_wait_asynccnt 0" ::: "memory");
}

// ---------------------------------------------------------------------------
// C[M,N] = A[M,K] * W[N,K]^T + bias[N]
// grid = (N/128, M/64), block = 256 (8 wave32s).
// Each wave computes a 16x64 strip via 4 accumulators of V_WMMA_F32_16X16X4_F32.
// K-tiles are double-buffered in LDS and staged with async-to-LDS copies that
// overlap the WMMA chain of the current tile.
// Requires: M % 64 == 0, N % 128 == 0, K % 32 == 0 (true for all call sites).
// ---------------------------------------------------------------------------
__global__ __launch_bounds__(256)
void gemm_awt_bias_kernel(const float* __restrict__ A, const float* __restrict__ W,
                          const float* __restrict__ bias, float* __restrict__ C,
                          int K, int ldc)
{
    __shared__ float As[2 * ASZ];
    __shared__ float Ws[2 * WSZ];

    const int tid  = threadIdx.x;
    const int wave = tid >> 5;
    const int lane = tid & 31;
    const int half = lane >> 4;   // 0: lanes 0-15, 1: lanes 16-31
    const int l15  = lane & 15;

    const long m_blk = (long)blockIdx.y * 64;
    const long n_blk = (long)blockIdx.x * 128;

    const int wm = (wave & 3) * 16;   // wave's M offset inside tile
    const int wn = (wave >> 2) * 64;  // wave's N offset inside tile

    v8f acc[4] = {};

    // Per-thread staging coordinates (each 16B chunk stays inside one LDS row,
    // so the padded LDP layout is preserved).
    int ar[2], ac[2], wr[4], wc[4];
    #pragma unroll
    for (int j = 0; j < 2; ++j) { int f4 = tid + j * 256; ar[j] = f4 >> 3; ac[j] = (f4 & 7) * 4; }
    #pragma unroll
    for (int j = 0; j < 4; ++j) { int f4 = tid + j * 256; wr[j] = f4 >> 3; wc[j] = (f4 & 7) * 4; }

    // ---- prologue: stage K-tile 0 into buffer 0 ----
    #pragma unroll
    for (int j = 0; j < 2; ++j)
        async_copy_b128(&As[ar[j] * LDP + ac[j]], A + (m_blk + ar[j]) * (long)K + ac[j]);
    #pragma unroll
    for (int j = 0; j < 4; ++j)
        async_copy_b128(&Ws[wr[j] * LDP + wc[j]], W + (n_blk + wr[j]) * (long)K + wc[j]);
    wait_async0();
    __syncthreads();

    int cur = 0;
    for (int k0 = 0; k0 < K; k0 += KT) {
        const int nxt = cur ^ 1;

        // ---- stage next K-tile into the other buffer (overlaps compute) ----
        if (k0 + KT < K) {
            const int kn = k0 + KT;
            #pragma unroll
            for (int j = 0; j < 2; ++j)
                async_copy_b128(&As[nxt * ASZ + ar[j] * LDP + ac[j]],
                                A + (m_blk + ar[j]) * (long)K + kn + ac[j]);
            #pragma unroll
            for (int j = 0; j < 4; ++j)
                async_copy_b128(&Ws[nxt * WSZ + wr[j] * LDP + wc[j]],
                                W + (n_blk + wr[j]) * (long)K + kn + wc[j]);
        }

        // ---- 8 k-steps of 4, 4 WMMAs each (A fragment reused across N) ----
        const float* Ab = &As[cur * ASZ];
        const float* Wb = &Ws[cur * WSZ];
        #pragma unroll
        for (int kk = 0; kk < KT; kk += 4) {
            const int kc = kk + 2 * half;  // lanes 0-15: K=kk,kk+1 ; 16-31: kk+2,kk+3
            v2f a;
            a.x = Ab[(wm + l15) * LDP + kc];
            a.y = Ab[(wm + l15) * LDP + kc + 1];
            #pragma unroll
            for (int j = 0; j < 4; ++j) {
                v2f b;
                b.x = Wb[(wn + j * 16 + l15) * LDP + kc];
                b.y = Wb[(wn + j * 16 + l15) * LDP + kc + 1];
                acc[j] = __builtin_amdgcn_wmma_f32_16x16x4_f32(
                    false, a, false, b, (short)0, acc[j], false, false);
            }
        }

        // next buffer filled (own asynccnt) + everyone done reading cur
        wait_async0();
        __syncthreads();
        cur = nxt;
    }

    // ---- epilogue: C/D layout is VGPR r -> M = r (lanes 0-15) / r+8 (16-31) ----
    #pragma unroll
    for (int j = 0; j < 4; ++j) {
        const long  ncol = n_blk + wn + j * 16 + l15;
        const float bv   = bias[ncol];
        #pragma unroll
        for (int r = 0; r < 8; ++r) {
            const long mrow = m_blk + wm + half * 8 + r;
            C[mrow * (long)ldc + ncol] = acc[j][r] + bv;
        }
    }
}

// ---------------------------------------------------------------------------
// Bahdanau attention for one step: score = Va . tanh(q + Uk) + Va_b,
// softmax over S, ctx = w @ enc.  One block per batch row.
// attn_out base already includes the t*S offset; row stride is T*S.
// ---------------------------------------------------------------------------
__global__ __launch_bounds__(256)
void attention_kernel(const float* __restrict__ q, const float* __restrict__ Uk,
                      const float* __restrict__ enc,
                      const float* __restrict__ Va_w, const float* __restrict__ Va_b,
                      float* __restrict__ ctx, float* __restrict__ attn_out)
{
    const int b    = blockIdx.x;
    const int tid  = threadIdx.x;
    const int wave = tid >> 5;
    const int lane = tid & 31;

    __shared__ float w_s[Ss];

    const float* qb  = q  + (long)b * Hh;
    const float* Ukb = Uk + (long)b * Ss * Hh;

    // scores: each wave covers 16 of the 128 source positions
    for (int s = wave; s < Ss; s += 8) {
        const float* u = Ukb + (long)s * Hh;
        float p = 0.f;
        for (int h = lane; h < Hh; h += 32)
            p += Va_w[h] * tanhf(qb[h] + u[h]);
        for (int off = 16; off; off >>= 1) p += __shfl_xor(p, off);
        if (lane == 0) w_s[s] = p + Va_b[0];
    }
    __syncthreads();

    // softmax over 128 entries (wave 0 only, shuffle reductions)
    if (wave == 0) {
        float m = -INFINITY;
        for (int s = lane; s < Ss; s += 32) m = fmaxf(m, w_s[s]);
        for (int off = 16; off; off >>= 1) m = fmaxf(m, __shfl_xor(m, off));
        float sum = 0.f;
        for (int s = lane; s < Ss; s += 32) { float e = __expf(w_s[s] - m); w_s[s] = e; sum += e; }
        for (int off = 16; off; off >>= 1) sum += __shfl_xor(sum, off);
        const float inv = 1.f / sum;
        for (int s = lane; s < Ss; s += 32) w_s[s] *= inv;
    }
    __syncthreads();

    for (int s = tid; s < Ss; s += 256)
        attn_out[(long)b * Tt * Ss + s] = w_s[s];

    const float* eb = enc + (long)b * Ss * Hh;
    for (int h = tid; h < Hh; h += 256) {
        float a = 0.f;
        for (int s = 0; s < Ss; ++s) a += w_s[s] * eb[(long)s * Hh + h];
        ctx[(long)b * Hh + h] = a;
    }
}

// x[b, 0:H] = emb_W[token(b,t)], x[b, H:2H] = ctx[b]
__global__ void embed_concat_kernel(const int* __restrict__ target,
                                    const float* __restrict__ emb_W,
                                    const float* __restrict__ ctx,
                                    float* __restrict__ x, int t)
{
    const int i = blockIdx.x * blockDim.x + threadIdx.x;
    if (i >= Bb * 2 * Hh) return;
    const int b = i / (2 * Hh);
    const int c = i % (2 * Hh);
    if (c < Hh) {
        const int tok = (t == 0) ? SOS : target[b * Tt + (t - 1)];
        x[i] = emb_W[(long)tok * Hh + c];
    } else {
        x[i] = ctx[(long)b * Hh + (c - Hh)];
    }
}

// GRU cell combine
__global__ void gru_combine_kernel(const float* __restrict__ gi, const float* __restrict__ gh,
                                   const float* __restrict__ h, float* __restrict__ h_new)
{
    const int i = blockIdx.x * blockDim.x + threadIdx.x;
    if (i >= Bb * Hh) return;
    const int b  = i / Hh;
    const int hh = i % Hh;
    const float* gib = gi + (long)b * 3 * Hh;
    const float* ghb = gh + (long)b * 3 * Hh;
    const float r = 1.f / (1.f + __expf(-(gib[hh] + ghb[hh])));
    const float z = 1.f / (1.f + __expf(-(gib[Hh + hh] + ghb[Hh + hh])));
    const float n = tanhf(gib[2 * Hh + hh] + r * ghb[2 * Hh + hh]);
    h_new[i] = (1.f - z) * n + z * h[i];
}

// in-place log_softmax over V=32000; one block per batch row
__global__ __launch_bounds__(256)
void log_softmax_kernel(float* __restrict__ out, int t)
{
    const int b    = blockIdx.x;
    const int tid  = threadIdx.x;
    const int lane = tid & 31;
    const int wave = tid >> 5;
    float* row = out + (long)b * Tt * Vv + (long)t * Vv;

    __shared__ float red_m[8];
    __shared__ float red_s[8];

    float m = -INFINITY;
    for (int v = tid; v < Vv; v += 256) m = fmaxf(m, row[v]);
    for (int off = 16; off; off >>= 1) m = fmaxf(m, __shfl_xor(m, off));
    if (lane == 0) red_m[wave] = m;
    __syncthreads();
    m = red_m[0];
    #pragma unroll
    for (int i = 1; i < 8; ++i) m = fmaxf(m, red_m[i]);

    float s = 0.f;
    for (int v = tid; v < Vv; v += 256) s += __expf(row[v] - m);
    for (int off = 16; off; off >>= 1) s += __shfl_xor(s, off);
    if (lane == 0) red_s[wave] = s;
    __syncthreads();
    s = 0.f;
    #pragma unroll
    for (int i = 0; i < 8; ++i) s += red_s[i];

    const float lse = m + __logf(s);
    for (int v = tid; v < Vv; v += 256) row[v] -= lse;
}

__global__ void copy_f32_kernel(float* __restrict__ dst, const float* __restrict__ src, int n)
{
    const int i = blockIdx.x * blockDim.x + threadIdx.x;
    if (i < n) dst[i] = src[i];
}

// ---------------------------------------------------------------------------
extern "C" void kernel_launch(void* const* d_in, const int* in_sizes, int n_in,
                              void* d_out, int out_size, void* d_ws, size_t ws_size,
                              hipStream_t stream)
{
    (void)in_sizes; (void)n_in; (void)out_size; (void)ws_size;

    const float* enc    = (const float*)d_in[0];   // [B,S,H]
    const float* hidden = (const float*)d_in[1];   // [1,B,H]
    const int*   target = (const int*)  d_in[2];   // [B,T]
    const float* emb_W  = (const float*)d_in[3];   // [V,H]
    const float* Wa_w   = (const float*)d_in[4];   // [H,H]
    const float* Wa_b   = (const float*)d_in[5];   // [H]
    const float* Ua_w   = (const float*)d_in[6];   // [H,H]
    const float* Ua_b   = (const float*)d_in[7];   // [H]
    const float* Va_w   = (const float*)d_in[8];   // [1,H]
    const float* Va_b   = (const float*)d_in[9];   // [1]
    const float* W_ih   = (const float*)d_in[10];  // [3H,2H]
    const float* W_hh   = (const float*)d_in[11];  // [3H,H]
    const float* b_ih   = (const float*)d_in[12];  // [3H]
    const float* b_hh   = (const float*)d_in[13];  // [3H]
    const float* out_W  = (const float*)d_in[14];  // [V,H]
    const float* out_b  = (const float*)d_in[15];  // [V]

    float* out      = (float*)d_out;
    float* out_logp = out;                               // [B,T,V]
    float* out_hT   = out + (long)Bb * Tt * Vv;          // [1,B,H]
    float* out_attn = out_hT + (long)Bb * Hh;            // [B,T,S]

    float* ws  = (float*)d_ws;
    float* Uk  = ws;  ws += (long)Bb * Ss * Hh;          // 33.5 MB
    float* h0  = ws;  ws += (long)Bb * Hh;
    float* h1  = ws;  ws += (long)Bb * Hh;
    float* q   = ws;  ws += (long)Bb * Hh;
    float* ctx = ws;  ws += (long)Bb * Hh;
    float* x   = ws;  ws += (long)Bb * 2 * Hh;
    float* gi  = ws;  ws += (long)Bb * 3 * Hh;
    float* gh  = ws;  ws += (long)Bb * 3 * Hh;

    // Uk[b,s,g] = enc[b,s,:] . Ua_w[g,:] + Ua_b[g]   (hoisted, constant per step)
    gemm_awt_bias_kernel<<<dim3(Hh / 128, (Bb * Ss) / 64), 256, 0, stream>>>(
        enc, Ua_w, Ua_b, Uk, Hh, Hh);

    copy_f32_kernel<<<(Bb * Hh + 255) / 256, 256, 0, stream>>>(h0, hidden, Bb * Hh);

    float* hcur = h0;
    float* hnext = h1;
    for (int t = 0; t < Tt; ++t) {
        // q = h @ Wa_w^T + Wa_b
        gemm_awt_bias_kernel<<<dim3(Hh / 128, 1), 256, 0, stream>>>(
            hcur, Wa_w, Wa_b, q, Hh, Hh);
        // attention: scores, softmax, ctx; writes attns[:, t, :]
        attention_kernel<<<Bb, 256, 0, stream>>>(
            q, Uk, enc, Va_w, Va_b, ctx, out_attn + (long)t * Ss);
        // x = [emb(token_t) ; ctx]
        embed_concat_kernel<<<(Bb * 2 * Hh + 255) / 256, 256, 0, stream>>>(
            target, emb_W, ctx, x, t);
        // gi = x @ W_ih^T + b_ih   (K = 2H)
        gemm_awt_bias_kernel<<<dim3((3 * Hh) / 128, 1), 256, 0, stream>>>(
            x, W_ih, b_ih, gi, 2 * Hh, 3 * Hh);
        // gh = h @ W_hh^T + b_hh   (K = H)
        gemm_awt_bias_kernel<<<dim3((3 * Hh) / 128, 1), 256, 0, stream>>>(
            hcur, W_hh, b_hh, gh, Hh, 3 * Hh);
        // GRU combine -> h_new
        gru_combine_kernel<<<(Bb * Hh + 255) / 256, 256, 0, stream>>>(gi, gh, hcur, hnext);
        // logits straight into d_out[b, t, :]  (row stride T*V)
        gemm_awt_bias_kernel<<<dim3(Vv / 128, 1), 256, 0, stream>>>(
            hnext, out_W, out_b, out_logp + (long)t * Vv, Hh, Tt * Vv);
        // in-place log_softmax over V
        log_softmax_kernel<<<Bb, 256, 0, stream>>>(out_logp, t);

        float* tmp = hcur; hcur = hnext; hnext = tmp;
    }

    copy_f32_kernel<<<(Bb * Hh + 255) / 256, 256, 0, stream>>>(out_hT, hcur, Bb * Hh);
}